// GuidedFilter_42460046688467
// MI455X (gfx1250) — compile-verified
//
#include <hip/hip_runtime.h>

#define HH 2048
#define WW 2048
#define LDSW 80   // padded LDS row stride (floats): 72 used, 80 avoids conflicts

typedef __attribute__((ext_vector_type(2))) float v2f;
typedef __attribute__((ext_vector_type(4))) float v4f;
typedef __attribute__((ext_vector_type(8))) float v8f;
typedef int v4i_g __attribute__((vector_size(16)));              // gcc-style, per diag
typedef __attribute__((address_space(1))) v4i_g as1_v4i;         // global int4
typedef __attribute__((address_space(3))) v4i_g as3_v4i;         // LDS int4

#if defined(__has_builtin)
# if __has_builtin(__builtin_amdgcn_global_load_async_to_lds_b128) && \
     __has_builtin(__builtin_amdgcn_s_wait_asynccnt)
#  define ASYNC_LDS 1
# endif
#endif
#ifndef ASYNC_LDS
# define ASYNC_LDS 0
#endif

// Line kernels as (column offset, vertical run d0..d1) groups, from
// round(d * tan(A)), d in [-24,24], A in {-10, 0, +10} deg.
constexpr int NG[3]      = {9, 1, 9};
constexpr int OFFT[3][9] = {
  { 4, 3, 2, 1, 0,-1,-2,-3,-4},
  { 0, 0, 0, 0, 0, 0, 0, 0, 0},
  {-4,-3,-2,-1, 0, 1, 2, 3, 4},
};
constexpr int D0T[3][9] = {
  {-24,-19,-14,-8,-2, 3, 9,15,20},
  {-24,  1,  1, 1, 1, 1, 1, 1, 1},
  {-24,-19,-14,-8,-2, 3, 9,15,20},
};
constexpr int D1T[3][9] = {
  {-20,-15,-9,-3, 2, 8,14,19,24},
  { 24,  0,  0, 0, 0, 0, 0, 0, 0},
  {-20,-15,-9,-3, 2, 8,14,19,24},
};

// One 128-thread block = 4 waves; wave w computes the 16x16 tile at
// (i0, j0b + 16w) of:  out = conv_KID(inA [- inB]) / N  [+ addp].
// Staging: the block's 64x72 input window (zero-padded, difference already
// applied) is placed in LDS once -- via the gfx1250 async-to-LDS engine on
// interior blocks of the non-subtract passes -- then each 49/16-tap line conv
// runs as banded f32 WMMA over K=4 slabs (0/1 weights => exact f32).
template <int KID, bool SUB, bool ADD>
__global__ void __launch_bounds__(128)
conv_pass(const float* __restrict__ inA, const float* __restrict__ inB,
          const float* __restrict__ addp, float* __restrict__ outp)
{
  __shared__ float tile[64 * LDSW];   // 20 KB

  const int tid  = threadIdx.x;
  const int lane = tid & 31;
  const int wv   = tid >> 5;
  const int half = lane >> 4;
  const int l15  = lane & 15;
  const int j0b  = blockIdx.x * 64;        // block column base
  const int i0   = blockIdx.y * 16;        // tile row base
  const int rTop = i0 - 24;                // window row 0 (abs)
  const int cLft = j0b - 4;                // window col 0 (abs)

  const bool interior = (i0 >= 24) && (i0 + 39 < HH) && (j0b >= 4) && (j0b + 67 < WW);

  if (interior) {
    // 64 rows x 18 float4 chunks = 1152 vec4 = 9 per thread, fully in-bounds.
    const float* baseA = inA + rTop * WW + cLft;
    const float* baseB = SUB ? (inB + rTop * WW + cLft) : nullptr;
#pragma unroll
    for (int k = 0; k < 9; ++k) {
      const int idx = tid + 128 * k;
      const int r = idx / 18, c4 = idx % 18;
      float* lp = &tile[r * LDSW + c4 * 4];
      if (SUB) {
        v4f a = *(const v4f*)(baseA + r * WW + c4 * 4);
        v4f b = *(const v4f*)(baseB + r * WW + c4 * 4);
        *(v4f*)lp = a - b;
      } else {
#if ASYNC_LDS
        __builtin_amdgcn_global_load_async_to_lds_b128(
            (as1_v4i*)(uintptr_t)(baseA + r * WW + c4 * 4),
            (as3_v4i*)(uint32_t)(uintptr_t)lp,
            0, 0);
#else
        *(v4f*)lp = *(const v4f*)(baseA + r * WW + c4 * 4);
#endif
      }
    }
#if ASYNC_LDS
    if (!SUB) __builtin_amdgcn_s_wait_asynccnt(0);
#endif
  } else {
    // Edge blocks: scalar staging with clamp + zero padding. 4608/128 = 36.
#pragma unroll 4
    for (int k = 0; k < 36; ++k) {
      const int idx = tid + 128 * k;
      const int r = idx / 72, c = idx % 72;
      const int ra = rTop + r, ca = cLft + c;
      const bool ok = ((unsigned)ra < (unsigned)HH) && ((unsigned)ca < (unsigned)WW);
      const int p = ok ? (ra * WW + ca) : 0;
      float v = inA[p];
      if (SUB) v -= inB[p];
      tile[r * LDSW + c] = ok ? v : 0.0f;
    }
  }
  __syncthreads();

  v8f acc = {};
#pragma unroll
  for (int g = 0; g < NG[KID]; ++g) {
    const int off = OFFT[KID][g];
    const int d0  = D0T[KID][g];
    const int d1  = D1T[KID][g];
    const int nb  = (d1 - d0 + 19) >> 2;           // K=4 slabs over the band
    const int wcol = wv * 16 + off + 4 + l15;      // LDS column for B
#pragma unroll
    for (int b = 0; b < nb; ++b) {
      const int s0 = d0 + 24 + b * 4;              // window row of K=0
      v2f A, Bv;
#pragma unroll
      for (int v = 0; v < 2; ++v) {
        const int K = half * 2 + v;                // 16x16x4 f32 K mapping
        const int c = s0 + K - 24;                 // band test: c - M in [d0,d1]
        A[v]  = ((unsigned)(l15 - (c - d1)) <= (unsigned)(d1 - d0)) ? 1.0f : 0.0f;
        Bv[v] = tile[(s0 + K) * LDSW + wcol];
      }
      acc = __builtin_amdgcn_wmma_f32_16x16x4_f32(
          false, A, false, Bv, (short)0, acc, false, false);
    }
  }

  // D layout: lane holds col j0+l15, rows i0 + v + 8*half.
  const int j = j0b + wv * 16 + l15;
#pragma unroll
  for (int v = 0; v < 8; ++v) {
    const int i = i0 + v + 8 * half;
    float Nc = 0.0f;                               // boxfilter(ones), closed form
#pragma unroll
    for (int g = 0; g < NG[KID]; ++g) {
      if ((unsigned)(j + OFFT[KID][g]) < (unsigned)WW) {
        int lo = i + D0T[KID][g]; if (lo < 0) lo = 0;
        int hi = i + D1T[KID][g]; if (hi > HH - 1) hi = HH - 1;
        int c = hi - lo + 1; if (c < 0) c = 0;
        Nc += (float)c;
      }
    }
    float val = acc[v] / Nc;                       // Nc >= 3 always
    if (ADD) val += addp[i * WW + j];
    outp[i * WW + j] = val;
  }
}

__device__ __forceinline__ float reflectf(float x, int size) {
  float span = (float)(size - 1);
  float ax = fabsf(x);
  float extra = fmodf(ax, span);
  float flips = floorf(ax / span);
  float out = (fmodf(flips, 2.0f) == 0.0f) ? extra : span - extra;
  return fminf(fmaxf(out, 0.0f), span);
}

// out = grid_sample(Xc - X, coor, reflect, bilinear) + hX
__global__ void __launch_bounds__(256)
grid_sample_kernel(const float* __restrict__ Xc, const float* __restrict__ X,
                   const float* __restrict__ coor, const float* __restrict__ hX,
                   float* __restrict__ out)
{
  int idx = blockIdx.x * 256 + threadIdx.x;
  if (idx >= HH * WW) return;
  float cx = coor[2 * idx + 0];
  float cy = coor[2 * idx + 1];
  float gx = reflectf((cx + 1.0f) * 0.5f * (float)(WW - 1), WW);
  float gy = reflectf((cy + 1.0f) * 0.5f * (float)(HH - 1), HH);
  float x0f = floorf(gx), y0f = floorf(gy);
  float wx = gx - x0f, wy = gy - y0f;
  int xi0 = (int)fminf(fmaxf(x0f,        0.0f), (float)(WW - 1));
  int yi0 = (int)fminf(fmaxf(y0f,        0.0f), (float)(HH - 1));
  int xi1 = (int)fminf(fmaxf(x0f + 1.0f, 0.0f), (float)(WW - 1));
  int yi1 = (int)fminf(fmaxf(y0f + 1.0f, 0.0f), (float)(HH - 1));
  int p00 = yi0 * WW + xi0, p01 = yi0 * WW + xi1;
  int p10 = yi1 * WW + xi0, p11 = yi1 * WW + xi1;
  float v00 = Xc[p00] - X[p00];
  float v01 = Xc[p01] - X[p01];
  float v10 = Xc[p10] - X[p10];
  float v11 = Xc[p11] - X[p11];
  float r = v00 * (1.0f - wx) * (1.0f - wy) + v01 * wx * (1.0f - wy)
          + v10 * (1.0f - wx) * wy          + v11 * wx * wy;
  out[idx] = r + hX[idx];
}

extern "C" void kernel_launch(void* const* d_in, const int* in_sizes, int n_in,
                              void* d_out, int out_size, void* d_ws, size_t ws_size,
                              hipStream_t stream) {
  const float* X    = (const float*)d_in[0];
  const float* y    = (const float*)d_in[1];
  const float* hX   = (const float*)d_in[2];
  const float* coor = (const float*)d_in[3];
  float* out = (float*)d_out;
  float* Xc  = (float*)d_ws;   // 16 MB persistent state across passes
  float* T   = out;            // d_out doubles as the ping temp until the end

  dim3 cg(WW / 64, HH / 16);
  dim3 cb(128);

  // kernel 0: T = conv0(y - X)/N0 ; Xc = X + conv0(T)/N0
  conv_pass<0, true,  false><<<cg, cb, 0, stream>>>(y, X,  nullptr, T);
  conv_pass<0, false, true ><<<cg, cb, 0, stream>>>(T, nullptr, X,  Xc);
  // kernel 1: T = conv1(y - Xc)/N1 ; Xc += conv1(T)/N1
  conv_pass<1, true,  false><<<cg, cb, 0, stream>>>(y, Xc, nullptr, T);
  conv_pass<1, false, true ><<<cg, cb, 0, stream>>>(T, nullptr, Xc, Xc);
  // kernel 2: T = conv2(y - Xc)/N2 ; Xc += conv2(T)/N2
  conv_pass<2, true,  false><<<cg, cb, 0, stream>>>(y, Xc, nullptr, T);
  conv_pass<2, false, true ><<<cg, cb, 0, stream>>>(T, nullptr, Xc, Xc);

  int n = HH * WW;
  grid_sample_kernel<<<(n + 255) / 256, 256, 0, stream>>>(Xc, X, coor, hX, out);
}